// MyModel_87522843560120
// MI455X (gfx1250) — compile-verified
//
#include <hip/hip_runtime.h>
#include <stdint.h>

// Reference: result = items (48,000,000 f32), final_pool = items[:1024] (12,288 f32).
// Pure memcpy: 192MB read + 192MB write -> ~16.5us floor at 23.3 TB/s HBM.
// Strategy: b128 per-lane accesses; on gfx1250 route the bulk copy through the
// CDNA5 async global<->LDS DMA path (ASYNCcnt), falling back to nontemporal
// vector copies if the async builtins are unavailable.

typedef float f4  __attribute__((ext_vector_type(4)));
typedef int   v4i __attribute__((ext_vector_type(4)));

#define TPB 256
#define BATCH 8   // 8 x 16B per lane per phase -> 32KB LDS tile per block

#if defined(__AMDGCN__) && defined(__gfx1250__) && \
    __has_builtin(__builtin_amdgcn_global_load_async_to_lds_b128) && \
    __has_builtin(__builtin_amdgcn_global_store_async_from_lds_b128) && \
    __has_builtin(__builtin_amdgcn_s_wait_asynccnt)
#define USE_ASYNC_LDS 1
#else
#define USE_ASYNC_LDS 0
#endif

#if defined(__AMDGCN__)
typedef __attribute__((address_space(3))) v4i lds_v4i;
#endif

__global__ void bulk_copy_kernel(const float* __restrict__ inF,
                                 float* __restrict__ outF,
                                 long long n4) {
  const f4* __restrict__ in  = (const f4*)inF;
  f4* __restrict__       out = (f4*)outF;

#if USE_ASYNC_LDS
  __shared__ f4 stage[TPB * BATCH];
  const long long tileElems = (long long)TPB * BATCH;
  const int tid = threadIdx.x;

  for (long long base = (long long)blockIdx.x * tileElems; base < n4;
       base += (long long)gridDim.x * tileElems) {
    // Phase 1: async DMA global -> LDS (per-lane b128), no VGPR round trip.
#pragma unroll
    for (int b = 0; b < BATCH; ++b) {
      long long g = base + (long long)b * TPB + tid;
      if (g < n4) {
        __builtin_amdgcn_global_load_async_to_lds_b128(
            (v4i*)(&in[g]),
            (lds_v4i*)(&stage[b * TPB + tid]),
            /*offset=*/0, /*cpol=*/0);
      }
    }
    // Async loads complete in order among themselves; drain them all.
    __builtin_amdgcn_s_wait_asynccnt(0);

    // Phase 2: async DMA LDS -> global. Each lane stores the slots it loaded,
    // so no workgroup barrier is required.
#pragma unroll
    for (int b = 0; b < BATCH; ++b) {
      long long g = base + (long long)b * TPB + tid;
      if (g < n4) {
        __builtin_amdgcn_global_store_async_from_lds_b128(
            (v4i*)(&out[g]),
            (lds_v4i*)(&stage[b * TPB + tid]),
            /*offset=*/0, /*cpol=*/0);
      }
    }
    // LDS tile is reused next iteration: wait for stores to drain.
    __builtin_amdgcn_s_wait_asynccnt(0);
  }
#else
  // Fallback: nontemporal b128 grid-stride copy (streaming data, keep L2 clean).
  long long tid    = (long long)blockIdx.x * blockDim.x + threadIdx.x;
  long long stride = (long long)gridDim.x * blockDim.x;
  for (long long i = tid; i < n4; i += stride) {
#if defined(__AMDGCN__)
    if (i + stride < n4) __builtin_prefetch(&in[i + stride], 0, 1);
#endif
    f4 v = __builtin_nontemporal_load(&in[i]);
    __builtin_nontemporal_store(v, &out[i]);
  }
#endif
}

// Copies the sampled pool (first pool_n floats of `in`) to out[n ..] and any
// scalar remainder of the bulk copy (n not divisible by 4; defensively).
__global__ void tail_copy_kernel(const float* __restrict__ in,
                                 float* __restrict__ out,
                                 long long n, long long n4, long long pool_n) {
  long long i = (long long)blockIdx.x * blockDim.x + threadIdx.x;
  if (i < pool_n) out[n + i] = in[i];
  long long rs  = n4 * 4;
  long long rem = n - rs;
  if (i < rem) out[rs + i] = in[rs + i];
}

extern "C" void kernel_launch(void* const* d_in, const int* in_sizes, int n_in,
                              void* d_out, int out_size, void* d_ws, size_t ws_size,
                              hipStream_t stream) {
  (void)n_in; (void)d_ws; (void)ws_size;
  const float* in  = (const float*)d_in[0];
  float*       out = (float*)d_out;

  long long n      = (long long)in_sizes[0];          // 48,000,000
  long long n4     = n / 4;                           // 12,000,000 float4
  long long pool_n = (long long)out_size - n;         // 12,288 floats
  if (pool_n < 0) pool_n = 0;

  // Bulk copy: grid-stride; 4096 blocks x 256 threads gives ample waves/WGP
  // (async path uses 32KB LDS per block -> ~10 resident blocks per WGP, so
  // plenty of outstanding async DMA to saturate 23.3 TB/s).
  bulk_copy_kernel<<<4096, TPB, 0, stream>>>(in, out, n4);

  long long rem   = n - n4 * 4;
  long long tailN = (pool_n > rem) ? pool_n : rem;
  if (tailN > 0) {
    int tb = (int)((tailN + TPB - 1) / TPB);
    tail_copy_kernel<<<tb, TPB, 0, stream>>>(in, out, n, n4, pool_n);
  }
}